// PiecewiseLinearShapeNN2D_79774722556361
// MI455X (gfx1250) — compile-verified
//
#include <hip/hip_runtime.h>

// MLS / RKPM shape-function evaluation on gfx1250 (MI455X).
// One wave32 per evaluation point m. Per element node (3x):
//   - 12-point patch pairwise Gram via V_WMMA_F32_16X16X4_F32
//   - assemble 15x17 augmented saddle system in LDS (16x17 tile, row 15 =
//     scratch so the R scatter is branchless)
//   - wave-cooperative Gauss-Jordan: lane i owns row i, pivot rows
//     broadcast via v_readlane_b32 (k is compile-time after unroll)
//   - contraction + butterfly reduction via ds_bpermute
// Fast paths: raw v_sqrt_f32, v_rcp_f32 + 1 Newton step for divides.

#define ALPHA_INV 5.0f        // 1/0.2
#define EPS_DIAG_F 1e-8f
#define EPS_R_F 1e-12f
#define NPATCH 12
#define NSYS 15               // 12 + M_POLY
#define NCOLS 17              // 15 + 2 rhs columns
#define LROWS 16              // 15 system rows + 1 scratch row
#define WAVES_PER_BLOCK 8

typedef __attribute__((ext_vector_type(2))) float v2f;
typedef __attribute__((ext_vector_type(8))) float v8f;

__device__ __forceinline__ float rlane(float v, int l) {
  return __builtin_bit_cast(float, __builtin_amdgcn_readlane(__builtin_bit_cast(int, v), l));
}
__device__ __forceinline__ float bperm(float v, int srcLane) {
  return __builtin_bit_cast(float, __builtin_amdgcn_ds_bpermute(srcLane << 2, __builtin_bit_cast(int, v)));
}
__device__ __forceinline__ float wave_sum(float v, int lane) {
#pragma unroll
  for (int s = 16; s >= 1; s >>= 1) v += bperm(v, lane ^ s);
  return v;
}
// v_rcp_f32 + one Newton-Raphson step (~full fp32 precision, no IEEE div seq)
__device__ __forceinline__ float frcp(float x) {
  float r = __builtin_amdgcn_rcpf(x);
  return r * (2.0f - x * r);
}
__device__ __forceinline__ float fsqrt(float x) {
  return __builtin_amdgcn_sqrtf(x);  // raw v_sqrt_f32 (~1 ulp)
}

__global__ __launch_bounds__(256) void mls_eval_kernel(
    const float* __restrict__ node_coords,  // (NNODES,2)
    const float* __restrict__ u,            // (NNODES,2)
    const float* __restrict__ x_eval,       // (MEVAL,2)
    const int*   __restrict__ conn,         // (NELEMS,3)
    const int*   __restrict__ patch,        // (NNODES,NPATCH)
    const int*   __restrict__ elem_id,      // (MEVAL,)
    float* __restrict__ out, int meval) {
  __shared__ float smem[WAVES_PER_BLOCK][LROWS * NCOLS];
  const int lane = threadIdx.x & 31;
  const int wid  = threadIdx.x >> 5;
  const int m0   = blockIdx.x * WAVES_PER_BLOCK + wid;
  const bool live = (m0 < meval);
  const int m = live ? m0 : (meval - 1);
  float* Asys = smem[wid];

  // ---- per-point geometry (all lanes redundantly; uniform addresses) ----
  const int eid = elem_id[m];
  const int n0 = conn[3 * eid + 0];
  const int n1 = conn[3 * eid + 1];
  const int n2 = conn[3 * eid + 2];
  const float v0x = node_coords[2 * n0], v0y = node_coords[2 * n0 + 1];
  const float v1x = node_coords[2 * n1], v1y = node_coords[2 * n1 + 1];
  const float v2x = node_coords[2 * n2], v2y = node_coords[2 * n2 + 1];
  const float xi = x_eval[2 * m], eta = x_eval[2 * m + 1];
  const float Nw[3] = {xi, eta, 1.0f - xi - eta};
  // Jmat columns are (v0-v2),(v1-v2)
  const float Ja = v0x - v2x, Jb = v1x - v2x;
  const float Jc = v0y - v2y, Jd = v1y - v2y;
  const float detJ = Ja * Jd - Jb * Jc;
  const float idet = frcp(detJ);
  const float Ji00 = Jd * idet, Ji01 = -Jb * idet;
  const float Ji10 = -Jc * idet, Ji11 = Ja * idet;
  // dN_dx[k][e] = Jinv @ [[1,0,-1],[0,1,-1]]
  const float dNdx0[3] = {Ji00, Ji01, -(Ji00 + Ji01)};
  const float dNdx1[3] = {Ji10, Ji11, -(Ji10 + Ji11)};
  const float xp = Nw[0] * v0x + Nw[1] * v1x + Nw[2] * v2x;
  const float yp = Nw[0] * v0y + Nw[1] * v1y + Nw[2] * v2y;

  const int nodes3[3] = {n0, n1, n2};
  float uhx = 0.f, uhy = 0.f;
  float g00 = 0.f, g01 = 0.f, g10 = 0.f, g11 = 0.f;

#pragma unroll
  for (int e = 0; e < 3; ++e) {
    const int node = nodes3[e];
    // lane j < 12 owns patch node j
    float cx = 0.f, cy = 0.f, ux = 0.f, uy = 0.f, pm = 0.f;
    if (lane < NPATCH) {
      const int p = patch[node * NPATCH + lane];
      pm = (p >= 0) ? 1.0f : 0.0f;
      const int ps = (p >= 0) ? p : 0;
      cx = node_coords[2 * ps];
      cy = node_coords[2 * ps + 1];
      ux = u[2 * ps];
      uy = u[2 * ps + 1];
    }
    const float nrm = cx * cx + cy * cy;  // |x_j|^2 (0 for lane>=12)

    // ---- WMMA Gram: D(16x16) = X(16x4) * X^T(4x16), K=0:x K=1:y K=2,3:0 ----
    // A layout (32b A 16x4): lanes0-15 VGPR0=K0,VGPR1=K1; lanes16-31 K2,K3.
    // cx,cy are zero for lanes>=12, so lanes16-31 carry the required zeros.
    v2f Am;
    Am[0] = cx;
    Am[1] = cy;
    v8f Cm = {0.f, 0.f, 0.f, 0.f, 0.f, 0.f, 0.f, 0.f};
    Cm = __builtin_amdgcn_wmma_f32_16x16x4_f32(false, Am, false, Am,
                                               (short)0, Cm, false, false);

    // D layout: lane L, VGPR v -> M = v + 8*(L>=16), N = L&15
    const int Ncol = lane & 15;
    const int hiOff = (lane >= 16) ? 8 : 0;
    const float nN = bperm(nrm, Ncol);
    const float vN = bperm(pm, Ncol);
#pragma unroll
    for (int v = 0; v < 8; ++v) {
      const int M = v + hiOff;  // 0..15; row 15 and col 15 land in scratch
      const float nM = bperm(nrm, M);
      const float vM = bperm(pm, M);
      const float r2 = fmaxf(nM + nN - 2.0f * Cm[v], 0.0f);
      const float rr = fsqrt(r2);
      const float s = rr * ALPHA_INV;
      const float w = 1.0f - s;
      float Rv = ((s <= 1.0f) && (vM * vN > 0.0f)) ? (w * w * (1.0f + 2.0f * s))
                                                   : 0.0f;
      if (M == Ncol && M < NPATCH) Rv += EPS_DIAG_F * (1.0f - vM);
      Asys[M * NCOLS + Ncol] = Rv;  // branchless; col 15 overwritten below
    }
    // P blocks, rhs (in-order same-wave LDS stores overwrite col-15 spills)
    if (lane < NPATCH) {
      Asys[lane * NCOLS + 12] = pm;
      Asys[lane * NCOLS + 13] = pm * cx;
      Asys[lane * NCOLS + 14] = pm * cy;
      Asys[12 * NCOLS + lane] = pm;
      Asys[13 * NCOLS + lane] = pm * cx;
      Asys[14 * NCOLS + lane] = pm * cy;
      Asys[lane * NCOLS + 15] = pm * ux;
      Asys[lane * NCOLS + 16] = pm * uy;
    } else if (lane < NSYS) {
      Asys[lane * NCOLS + 15] = 0.0f;
      Asys[lane * NCOLS + 16] = 0.0f;
    }
    __syncthreads();

    // ---- row-owner load ----
    float rw[NCOLS];
#pragma unroll
    for (int j = 0; j < NCOLS; ++j) rw[j] = 0.0f;
    if (lane < NSYS) {
#pragma unroll
      for (int j = 0; j < NCOLS; ++j) rw[j] = Asys[lane * NCOLS + j];
    }
    __syncthreads();

    // ---- wave-cooperative Gauss-Jordan (pivot broadcast = v_readlane) ----
    float dg = 1.0f;
#pragma unroll
    for (int k = 0; k < NSYS; ++k) {
      const float pkk = rlane(rw[k], k);
      const float invp = frcp(pkk);
      const float f = (lane == k || lane >= NSYS) ? 0.0f : rw[k] * invp;
#pragma unroll
      for (int j = k + 1; j < NCOLS; ++j)
        rw[j] = fmaf(-f, rlane(rw[j], k), rw[j]);
      if (lane == k) dg = pkk;  // diagonal frozen after step k
    }
    const float invdg = frcp(dg);
    const float bxl = (lane < NSYS) ? rw[15] * invdg : 0.0f;
    const float byl = (lane < NSYS) ? rw[16] * invdg : 0.0f;

    // ---- RP / dRP per lane ----
    float RPv = 0.f, dRx = 0.f, dRy = 0.f;
    if (lane < NPATCH) {
      const float dvx = xp - cx, dvy = yp - cy;
      const float rr = fsqrt(dvx * dvx + dvy * dvy);
      const float s = rr * ALPHA_INV;
      const bool inside = (s <= 1.0f) && (pm > 0.0f);
      const float w = 1.0f - s;
      RPv = inside ? (w * w * (1.0f + 2.0f * s)) : 0.0f;
      const float dRdr = inside ? (-6.0f * w * ALPHA_INV) : 0.0f;  // matches ref
      const float ir = frcp(rr + EPS_R_F);
      dRx = dRdr * dvx * ir;
      dRy = dRdr * dvy * ir;
    } else if (lane == 12) {
      RPv = 1.0f;
    } else if (lane == 13) {
      RPv = xp; dRx = 1.0f;
    } else if (lane == 14) {
      RPv = yp; dRy = 1.0f;
    }

    const float wux = wave_sum(RPv * bxl, lane);
    const float wuy = wave_sum(RPv * byl, lane);
    const float d00 = wave_sum(dRx * bxl, lane);  // dWu[x][x]
    const float d01 = wave_sum(dRy * bxl, lane);  // dWu[x][y]
    const float d10 = wave_sum(dRx * byl, lane);  // dWu[y][x]
    const float d11 = wave_sum(dRy * byl, lane);  // dWu[y][y]

    const float Ne = Nw[e];
    uhx = fmaf(Ne, wux, uhx);
    uhy = fmaf(Ne, wuy, uhy);
    g00 += dNdx0[e] * wux + Ne * d00;
    g01 += dNdx1[e] * wux + Ne * d01;
    g10 += dNdx0[e] * wuy + Ne * d10;
    g11 += dNdx1[e] * wuy + Ne * d11;
  }

  if (live && lane == 0) {
    out[2 * m + 0] = uhx;
    out[2 * m + 1] = uhy;
    out[2 * meval + m] = detJ;
    float* go = out + 3 * meval + 4 * m;
    go[0] = g00;
    go[1] = g01;
    go[2] = g10;
    go[3] = g11;
  }
}

extern "C" void kernel_launch(void* const* d_in, const int* in_sizes, int n_in,
                              void* d_out, int out_size, void* d_ws,
                              size_t ws_size, hipStream_t stream) {
  const float* node_coords = (const float*)d_in[0];
  const float* u           = (const float*)d_in[1];
  const float* x_eval      = (const float*)d_in[2];
  const int*   conn        = (const int*)d_in[3];
  const int*   patch       = (const int*)d_in[4];
  const int*   elem        = (const int*)d_in[5];
  float* out = (float*)d_out;
  const int meval = in_sizes[5];
  const int blocks = (meval + WAVES_PER_BLOCK - 1) / WAVES_PER_BLOCK;
  mls_eval_kernel<<<blocks, WAVES_PER_BLOCK * 32, 0, stream>>>(
      node_coords, u, x_eval, conn, patch, elem, out, meval);
}